// ReferenceSampler_68221260529947
// MI455X (gfx1250) — compile-verified
//
#include <hip/hip_runtime.h>
#include <math.h>

// ---------------- problem constants ----------------
#define NQ     256          // 4*64 queries
#define NT     100000       // targets
#define DIM    128
#define KTOP   100
#define SEG    50           // target segments
#define TPS    125          // 16-target tiles per segment (125*16 = 2000)
#define NBT    252          // 4 * 63 (b,t) transition slices
#define LROW   132          // LDS row stride in dwords (128 + 4 TDM pad) -> conflict-free b64 reads

typedef float v2f  __attribute__((ext_vector_type(2)));
typedef float v8f  __attribute__((ext_vector_type(8)));
typedef unsigned int u32x4 __attribute__((ext_vector_type(4)));
typedef int   i32x8 __attribute__((ext_vector_type(8)));
typedef int   i32x4 __attribute__((ext_vector_type(4)));

__device__ __forceinline__ v8f wmma_f32_4(v2f a, v2f b, v8f c) {
    // D = A(16x4 f32) * B(4x16 f32) + C(16x16 f32)
    return __builtin_amdgcn_wmma_f32_16x16x4_f32(
        /*neg_a=*/false, a, /*neg_b=*/false, b,
        /*c_mod=*/(short)0, c, /*reuse_a=*/false, /*reuse_b=*/false);
}

// TDM 2-D tile load: `rows` rows of 128 f32, global row stride 128 elements,
// LDS rows padded to 132 dwords (pad_interval=128 DW, pad_amount=4 DW).
// Rows beyond `tensor_rows` read as zero (OOB semantics).
__device__ __forceinline__ void tdm_load_tile(unsigned lds_off, const float* gaddr,
                                              unsigned rows, unsigned tensor_rows) {
    unsigned long long ga = (unsigned long long)gaddr;
    u32x4 g0;
    g0[0] = 1u;                                            // count=1, user descriptor
    g0[1] = lds_off;                                       // lds_addr (bytes)
    g0[2] = (unsigned)(ga & 0xffffffffull);                // global_addr[31:0]
    g0[3] = (unsigned)((ga >> 32) & 0x1ffffffull)          // global_addr[56:32]
          | (2u << 30);                                    // type = 2 ("image")
    i32x8 g1;
    g1[0] = (int)((3u << 25)   // pad_amount  = 4 dwords
                | (6u << 22)   // pad_interval= 128 dwords
                | (1u << 20)   // pad_enable
                | (2u << 16)); // data_size = 4 bytes ; workgroup_mask = 0
    g1[1] = (int)(128u << 16);         // tensor_dim0 = 128 (bits 79:48 low half)
    g1[2] = (int)(tensor_rows << 16);  // tensor_dim1[15:0] (bits 95:80)
    g1[3] = (int)(128u << 16);         // tile_dim0 = 128 (bits 127:112)
    g1[4] = (int)rows;                 // tile_dim1 (bits 143:128)
    g1[5] = 128;                       // tensor_dim0_stride = 128 elements
    g1[6] = 0;
    g1[7] = 0;
    i32x4 z4 = {};
    i32x8 z8 = {};
    // clang-23 / therock-10.0 form: 6 args (groups 0..3, extra group, cpol)
    __builtin_amdgcn_tensor_load_to_lds(g0, g1, z4, z4, z8, 0);
}

__device__ __forceinline__ unsigned lds_addr_of(const void* p) {
    // flat LDS pointer: low 32 bits are the wave-relative LDS byte offset
    return (unsigned)(unsigned long long)p;
}

// ---------------- kernel 0: norms ----------------
__global__ void norms_kernel(const float* __restrict__ q,
                             const float* __restrict__ t,
                             float* __restrict__ qn,
                             float* __restrict__ tn) {
    int i = blockIdx.x * blockDim.x + threadIdx.x;
    if (i < NT) {
        const float* p = t + (size_t)i * DIM;
        float s = 0.f;
        #pragma unroll 8
        for (int k = 0; k < DIM; ++k) s += p[k] * p[k];
        tn[i] = s;
    } else if (i < NT + NQ) {
        int j = i - NT;
        const float* p = q + (size_t)j * DIM;
        float s = 0.f;
        #pragma unroll 8
        for (int k = 0; k < DIM; ++k) s += p[k] * p[k];
        qn[j] = s;
    }
}

// ---------------- kernel 1: TDM-staged distances (WMMA f32) + segment top-K ----------------
// grid: (16 query groups, SEG segments), block: 32 (one wave32)
__global__ void __launch_bounds__(32)
knn_topk_kernel(const float* __restrict__ q, const float* __restrict__ t,
                const float* __restrict__ tn, const float* __restrict__ qn,
                float* __restrict__ cand_d, int* __restrict__ cand_i) {
    const int qg   = blockIdx.x;       // 16 query rows each
    const int seg  = blockIdx.y;
    const int lane = threadIdx.x;      // 0..31
    const int row  = lane & 15;        // M (A) / N (B) index
    const int klo  = (lane >> 4) * 2;  // K pair selector per ISA A/B layout

    __shared__ __align__(16) float lds_q[16 * LROW];
    __shared__ __align__(16) float lds_t[2][16 * LROW];
    __shared__ float topd[16][KTOP + 4];
    __shared__ int   topi[16][KTOP + 4];
    __shared__ float dtile[16][17];

    // async: queries once + first target tile (double buffered)
    tdm_load_tile(lds_addr_of(lds_q),     q + (size_t)qg * 16 * DIM,          16, 16);
    tdm_load_tile(lds_addr_of(lds_t[0]),  t + (size_t)(seg * TPS) * 16 * DIM, 16, 16);

    // this lane produces D rows M = r + 8*(lane>=16)
    float qnv[8];
    #pragma unroll
    for (int r = 0; r < 8; ++r) qnv[r] = qn[qg * 16 + r + 8 * (lane >> 4)];

    int   cnt  = 0;
    float wmax = -1e30f;
    int   wpos = 0;

    for (int tile = 0; tile < TPS; ++tile) {
        const int cur   = tile & 1;
        const int tbase = (seg * TPS + tile) * 16;

        if (tile + 1 < TPS) {   // overlap next tile's DMA with this tile's WMMAs
            tdm_load_tile(lds_addr_of(lds_t[1 - cur]),
                          t + (size_t)(tbase + 16) * DIM, 16, 16);
            __builtin_amdgcn_s_wait_tensorcnt(1);  // in-order: current tile done
        } else {
            __builtin_amdgcn_s_wait_tensorcnt(0);
        }
        asm volatile("" ::: "memory");  // keep LDS reads below the wait

        v8f acc = {};
        const float* ap = lds_q       + row * LROW + klo;
        const float* bp = &lds_t[cur][ row * LROW + klo];
        #pragma unroll 4
        for (int kb = 0; kb < DIM; kb += 4) {
            v2f a = *(const v2f*)(ap + kb);
            v2f b = *(const v2f*)(bp + kb);
            acc = wmma_f32_4(a, b, acc);
        }

        // sqdist tile -> LDS (each lane owns 8 rows, fixed column = row)
        float tnv = tn[tbase + row];
        #pragma unroll
        for (int r = 0; r < 8; ++r) {
            int M = r + 8 * (lane >> 4);
            dtile[M][row] = qnv[r] + tnv - 2.0f * acc[r];
        }
        __syncthreads();

        // lanes 0..15: replace-max top-K for query `lane`
        if (lane < 16) {
            #pragma unroll 1
            for (int c = 0; c < 16; ++c) {
                float d  = dtile[lane][c];
                int   gi = tbase + c;
                if (cnt < KTOP) {
                    topd[lane][cnt] = d; topi[lane][cnt] = gi;
                    if (d > wmax) { wmax = d; wpos = cnt; }
                    ++cnt;
                } else if (d < wmax) {
                    topd[lane][wpos] = d; topi[lane][wpos] = gi;
                    float m = topd[lane][0]; int p = 0;
                    #pragma unroll 4
                    for (int s = 1; s < KTOP; ++s) {
                        float v = topd[lane][s];
                        if (v > m) { m = v; p = s; }
                    }
                    wmax = m; wpos = p;
                }
            }
        }
        __syncthreads();
    }

    if (lane < 16) {
        size_t base = ((size_t)(qg * 16 + lane) * SEG + seg) * KTOP;
        for (int s = 0; s < KTOP; ++s) {
            cand_d[base + s] = topd[lane][s];
            cand_i[base + s] = topi[lane][s];
        }
    }
}

// ---------------- kernel 2: exact merge top-K + softmax + gather states ----------------
// grid: 256 (one per query), block: 256
__global__ void __launch_bounds__(256)
merge_kernel(const float* __restrict__ cand_d, const int* __restrict__ cand_i,
             const float* __restrict__ targets, const float* __restrict__ tn,
             float* __restrict__ emit, int* __restrict__ sidx,
             float* __restrict__ snorm, float* __restrict__ states) {
    const int qid = blockIdx.x;
    const int tid = threadIdx.x;
    const int NC  = SEG * KTOP;  // 5000 candidates

    __shared__ float cd[SEG * KTOP];
    __shared__ int   ci[SEG * KTOP];
    __shared__ float rmin[256];
    __shared__ int   rpos[256];
    __shared__ float seld[KTOP];
    __shared__ int   seli[KTOP];
    __shared__ float s_dmin, s_invsum;

    for (int k = tid; k < NC; k += 256) {
        cd[k] = cand_d[(size_t)qid * NC + k];
        ci[k] = cand_i[(size_t)qid * NC + k];
    }
    __syncthreads();

    for (int r = 0; r < KTOP; ++r) {
        float lm = 1e30f; int lp = 0;
        for (int k = tid; k < NC; k += 256)
            if (cd[k] < lm) { lm = cd[k]; lp = k; }
        rmin[tid] = lm; rpos[tid] = lp;
        __syncthreads();
        for (int s = 128; s > 0; s >>= 1) {
            if (tid < s && rmin[tid + s] < rmin[tid]) {
                rmin[tid] = rmin[tid + s]; rpos[tid] = rpos[tid + s];
            }
            __syncthreads();
        }
        if (tid == 0) {
            seld[r] = rmin[0];
            seli[r] = ci[rpos[0]];
            cd[rpos[0]] = 1e30f;   // remove winner
        }
        __syncthreads();
    }

    if (tid == 0) {  // max-shifted softmax over -distance (seld ascending)
        float dmin = seld[0], sum = 0.f;
        for (int k = 0; k < KTOP; ++k) sum += expf(dmin - seld[k]);
        s_dmin = dmin; s_invsum = 1.0f / sum;
    }
    __syncthreads();

    if (tid < KTOP) {
        emit [(size_t)qid * KTOP + tid] = expf(s_dmin - seld[tid]) * s_invsum;
        sidx [(size_t)qid * KTOP + tid] = seli[tid];
        snorm[(size_t)qid * KTOP + tid] = tn[seli[tid]];
    }
    // gather states[qid][e][:] = targets[seli[e]][:]
    for (int x = tid; x < KTOP * DIM; x += 256) {
        int e = x >> 7, c = x & (DIM - 1);
        states[(size_t)qid * KTOP * DIM + x] = targets[(size_t)seli[e] * DIM + c];
    }
}

// ---------------- kernel 3: TDM-staged transition probs (WMMA f32) ----------------
// grid: (252 (b,t) slices, 49 tiles), block: 32 (one wave32)
__global__ void __launch_bounds__(32)
trans_kernel(const float* __restrict__ states, const float* __restrict__ snorm,
             float* __restrict__ trans) {
    const int bt   = blockIdx.x;          // b*63 + t
    const int b    = bt / 63, tt = bt % 63;
    const int tij  = blockIdx.y;
    const int ti   = tij / 7, tj = tij % 7;
    const int lane = threadIdx.x;
    const int row  = lane & 15;
    const int klo  = (lane >> 4) * 2;

    const int qa = b * 64 + tt;       // a = states[b][t]
    const int qb = qa + 1;            // b = states[b][t+1]
    const int i0 = ti * 16, j0 = tj * 16;

    __shared__ __align__(16) float ldsA[16 * LROW];
    __shared__ __align__(16) float ldsB[16 * LROW];

    // rows beyond (KTOP - i0) come back as zero via TDM OOB semantics;
    // the i<100 / j<100 guard below never writes those lanes out.
    tdm_load_tile(lds_addr_of(ldsA), states + ((size_t)qa * KTOP + i0) * DIM,
                  16, (unsigned)(KTOP - i0));
    tdm_load_tile(lds_addr_of(ldsB), states + ((size_t)qb * KTOP + j0) * DIM,
                  16, (unsigned)(KTOP - j0));
    __builtin_amdgcn_s_wait_tensorcnt(0);
    asm volatile("" ::: "memory");

    v8f acc = {};
    const float* ap = ldsA + row * LROW + klo;
    const float* bp = ldsB + row * LROW + klo;
    #pragma unroll 4
    for (int kb = 0; kb < DIM; kb += 4) {
        v2f a  = *(const v2f*)(ap + kb);
        v2f bb = *(const v2f*)(bp + kb);
        acc = wmma_f32_4(a, bb, acc);
    }

    const int   j   = j0 + row;                                   // lane's N column
    const float bnj = (j < KTOP) ? snorm[(size_t)qb * KTOP + j] : 0.f;
    #pragma unroll
    for (int r = 0; r < 8; ++r) {
        int i = i0 + r + 8 * (lane >> 4);
        if (i < KTOP && j < KTOP) {
            float an = snorm[(size_t)qa * KTOP + i];
            float d  = an + bnj - 2.0f * acc[r];
            trans[((size_t)bt * KTOP + i) * KTOP + j] = expf(-d);
        }
    }
}

// ---------------- kernel 4: Viterbi scan + backtrace + output gather ----------------
// grid: 4 (batch), block: 128
__global__ void __launch_bounds__(128)
viterbi_kernel(const float* __restrict__ emit, const float* __restrict__ trans,
               const float* __restrict__ states, float* __restrict__ out) {
    const int b   = blockIdx.x;
    const int tid = threadIdx.x;

    __shared__ float v[KTOP], vn[KTOP];
    __shared__ int   back[63][KTOP];
    __shared__ float red[128];
    __shared__ int   path[64];

    if (tid < KTOP) v[tid] = emit[(size_t)(b * 64) * KTOP + tid];
    __syncthreads();

    for (int t = 0; t < 63; ++t) {
        const float* tr = trans + (size_t)(b * 63 + t) * KTOP * KTOP;
        if (tid < KTOP) {
            float best = -1e30f; int bi = 0;
            for (int i = 0; i < KTOP; ++i) {
                float s = v[i] * tr[(size_t)i * KTOP + tid];
                if (s > best) { best = s; bi = i; }   // first-max, matches argmax
            }
            back[t][tid] = bi;
            vn[tid] = best * emit[(size_t)(b * 64 + t + 1) * KTOP + tid];
        }
        __syncthreads();
        red[tid] = (tid < KTOP) ? vn[tid] : -1e30f;
        __syncthreads();
        for (int s = 64; s > 0; s >>= 1) {
            if (tid < s) red[tid] = fmaxf(red[tid], red[tid + s]);
            __syncthreads();
        }
        float inv = 1.0f / fmaxf(red[0], 1e-30f);
        if (tid < KTOP) v[tid] = vn[tid] * inv;
        __syncthreads();
    }

    if (tid == 0) {
        float best = -1e30f; int bi = 0;
        for (int i = 0; i < KTOP; ++i)
            if (v[i] > best) { best = v[i]; bi = i; }
        path[63] = bi;
        for (int t = 62; t >= 0; --t) path[t] = back[t][path[t + 1]];
    }
    __syncthreads();

    for (int x = tid; x < 64 * DIM; x += 128) {
        int t = x >> 7, c = x & (DIM - 1);
        out[((size_t)(b * 64) + t) * DIM + c] =
            states[((size_t)(b * 64 + t) * KTOP + path[t]) * DIM + c];
    }
}

// ---------------- host launcher ----------------
extern "C" void kernel_launch(void* const* d_in, const int* in_sizes, int n_in,
                              void* d_out, int out_size, void* d_ws, size_t ws_size,
                              hipStream_t stream) {
    const float* q = (const float*)d_in[0];   // (4,64,128)
    const float* t = (const float*)d_in[1];   // (100000,128)
    float* out = (float*)d_out;               // (4,64,128)

    char*  ws  = (char*)d_ws;
    size_t off = 0;
    auto carve = [&](size_t bytes) -> void* {
        void* p = ws + off;
        off += (bytes + 255) & ~(size_t)255;
        return p;
    };
    float* tn     = (float*)carve((size_t)NT * 4);
    float* qn     = (float*)carve((size_t)NQ * 4);
    float* cand_d = (float*)carve((size_t)NQ * SEG * KTOP * 4);
    int*   cand_i = (int*)  carve((size_t)NQ * SEG * KTOP * 4);
    float* emit   = (float*)carve((size_t)NQ * KTOP * 4);
    int*   sidx   = (int*)  carve((size_t)NQ * KTOP * 4);
    float* snorm  = (float*)carve((size_t)NQ * KTOP * 4);
    float* states = (float*)carve((size_t)NQ * KTOP * DIM * 4);
    float* trans  = (float*)carve((size_t)NBT * KTOP * KTOP * 4);

    norms_kernel  <<<(NT + NQ + 255) / 256, 256, 0, stream>>>(q, t, qn, tn);
    knn_topk_kernel<<<dim3(16, SEG),        32,  0, stream>>>(q, t, tn, qn, cand_d, cand_i);
    merge_kernel  <<<NQ,                    256, 0, stream>>>(cand_d, cand_i, t, tn,
                                                              emit, sidx, snorm, states);
    trans_kernel  <<<dim3(NBT, 49),         32,  0, stream>>>(states, snorm, trans);
    viterbi_kernel<<<4,                     128, 0, stream>>>(emit, trans, states, out);
}